// ClustCNNNodeEncoder_1589137899698
// MI455X (gfx1250) — compile-verified
//
#include <hip/hip_runtime.h>
#include <stdint.h>

typedef __attribute__((ext_vector_type(4))) unsigned int u32x4;
typedef __attribute__((ext_vector_type(8))) int          i32x8;
typedef __attribute__((ext_vector_type(4))) int          i32x4;

#define GS       32               // grid size S
#define VOX      (GS * GS * GS)   // 32768 voxels per cluster
#define PPC      4096             // points per cluster
#define NTHREADS 512              // 16 wave32s
#define PPT      (PPC / NTHREADS) // 8 points per thread

__global__ __launch_bounds__(NTHREADS)
void clustcnn_voxelize_kernel(const float* __restrict__ data,
                              const int*   __restrict__ clusts,
                              float*       __restrict__ out)
{
    extern __shared__ float grid[];        // VOX floats (128 KB dynamic LDS)
    __shared__ float redmin[16][3];
    __shared__ float redmax[16][3];
    __shared__ float sfin[8];              // mins[3], ranges[3], grid_step, max_range

    const int tid = threadIdx.x;
    const int c   = blockIdx.x;

    // ---- Phase 0: zero the LDS voxel grid (float4 stores) ----
    float4* g4 = (float4*)grid;
    #pragma unroll
    for (int i = 0; i < VOX / 4 / NTHREADS; ++i)
        g4[tid + i * NTHREADS] = make_float4(0.f, 0.f, 0.f, 0.f);

    // ---- Phase 1: gather 8 points into registers, local min/max ----
    float px[PPT], py[PPT], pz[PPT], pv[PPT];
    float mnx = 3.4e38f, mny = 3.4e38f, mnz = 3.4e38f;
    float mxx = -3.4e38f, mxy = -3.4e38f, mxz = -3.4e38f;
    const int* crow = clusts + (size_t)c * PPC;
    #pragma unroll
    for (int k = 0; k < PPT; ++k) {
        int idx = crow[tid + k * NTHREADS];
        const float* r = data + (size_t)idx * 5;
        float x = r[0], y = r[1], z = r[2], v = r[4];
        px[k] = x; py[k] = y; pz[k] = z; pv[k] = v;
        mnx = fminf(mnx, x); mxx = fmaxf(mxx, x);
        mny = fminf(mny, y); mxy = fmaxf(mxy, y);
        mnz = fminf(mnz, z); mxz = fmaxf(mxz, z);
    }

    // ---- Phase 2: wave32 butterfly reduction of min/max ----
    #pragma unroll
    for (int off = 16; off > 0; off >>= 1) {
        mnx = fminf(mnx, __shfl_xor(mnx, off, 32));
        mny = fminf(mny, __shfl_xor(mny, off, 32));
        mnz = fminf(mnz, __shfl_xor(mnz, off, 32));
        mxx = fmaxf(mxx, __shfl_xor(mxx, off, 32));
        mxy = fmaxf(mxy, __shfl_xor(mxy, off, 32));
        mxz = fmaxf(mxz, __shfl_xor(mxz, off, 32));
    }
    const int wave = tid >> 5;
    if ((tid & 31) == 0) {
        redmin[wave][0] = mnx; redmin[wave][1] = mny; redmin[wave][2] = mnz;
        redmax[wave][0] = mxx; redmax[wave][1] = mxy; redmax[wave][2] = mxz;
    }
    __syncthreads();   // grid zeroed + partials visible

    if (tid == 0) {
        float a0 = redmin[0][0], a1 = redmin[0][1], a2 = redmin[0][2];
        float b0 = redmax[0][0], b1 = redmax[0][1], b2 = redmax[0][2];
        #pragma unroll
        for (int w = 1; w < 16; ++w) {
            a0 = fminf(a0, redmin[w][0]); a1 = fminf(a1, redmin[w][1]); a2 = fminf(a2, redmin[w][2]);
            b0 = fmaxf(b0, redmax[w][0]); b1 = fmaxf(b1, redmax[w][1]); b2 = fmaxf(b2, redmax[w][2]);
        }
        float r0 = b0 - a0, r1 = b1 - a1, r2 = b2 - a2;
        float mr = fmaxf(fmaxf(r0, r1), r2) + 1.0f;
        sfin[0] = a0; sfin[1] = a1; sfin[2] = a2;
        sfin[3] = r0; sfin[4] = r1; sfin[5] = r2;
        sfin[6] = 1.0f / mr;   // grid_step
        sfin[7] = mr;          // 1/grid_step
    }
    __syncthreads();

    const float m0 = sfin[0], m1 = sfin[1], m2 = sfin[2];
    const float r0 = sfin[3], r1 = sfin[4], r2 = sfin[5];
    const float gstep = sfin[6], inv_gstep = sfin[7];
    const float fS = (float)GS, stp = 1.0f / (float)GS;

    // ---- Phase 3: separable trilinear splat ----
    // Per point: 6 per-dimension (frac, valid, scaled-clamped-index) triples,
    // then 8 corners are just 2 mults + 2 int adds + ds_add_f32 each.
    #pragma unroll
    for (int k = 0; k < PPT; ++k) {
        float v0 = (px[k] - m0 - r0 * 0.5f - 0.5f) * gstep + 0.5f;
        float v1 = (py[k] - m1 - r1 * 0.5f - 0.5f) * gstep + 0.5f;
        float v2 = (pz[k] - m2 - r2 * 0.5f - 0.5f) * gstep + 0.5f;
        int i0 = (int)floorf(v0 * fS);
        int i1 = (int)floorf(v1 * fS);
        int i2 = (int)floorf(v2 * fS);
        int a0 = (int)floorf((v0 - 1e-9f + gstep) * fS) + 1;
        int a1 = (int)floorf((v1 - 1e-9f + gstep) * fS) + 1;
        int a2 = (int)floorf((v2 - 1e-9f + gstep) * fS) + 1;
        const float val = pv[k];
        const float h0 = v0 + gstep, h1 = v1 + gstep, h2 = v2 + gstep;

        float fx[2], fy[2], fz[2];
        bool  vx[2], vy[2], vz[2];
        int   ox[2], oy[2], oz[2];   // pre-scaled clamped indices
        #pragma unroll
        for (int t = 0; t < 2; ++t) {
            int ix = i0 + t, iy = i1 + t, iz = i2 + t;
            vx[t] = ix < a0; vy[t] = iy < a1; vz[t] = iz < a2;
            fx[t] = (fminf(h0, (float)(ix + 1) * stp) - fmaxf(v0, (float)ix * stp)) * inv_gstep;
            fy[t] = (fminf(h1, (float)(iy + 1) * stp) - fmaxf(v1, (float)iy * stp)) * inv_gstep;
            fz[t] = (fminf(h2, (float)(iz + 1) * stp) - fmaxf(v2, (float)iz * stp)) * inv_gstep;
            ox[t] = min(max(ix, 0), GS - 1) * (GS * GS);
            oy[t] = min(max(iy, 0), GS - 1) * GS;
            oz[t] = min(max(iz, 0), GS - 1);
        }
        // fold the value into the x-fractions (saves one mult per corner)
        fx[0] *= val; fx[1] *= val;

        #pragma unroll
        for (int o = 0; o < 8; ++o) {
            const int a = (o >> 2) & 1, b = (o >> 1) & 1, d = o & 1;
            if (vx[a] && vy[b] && vz[d])
                atomicAdd(&grid[ox[a] + oy[b] + oz[d]], fx[a] * fy[b] * fz[d]);
        }
    }
    __syncthreads();   // all ds_add_f32 complete across the workgroup

    // ---- Phase 4: bulk LDS -> global writeback via Tensor Data Mover ----
#if __has_builtin(__builtin_amdgcn_tensor_store_from_lds)
    if (tid == 0) {
        uint32_t lds_base = (uint32_t)(uintptr_t)(void*)grid;   // LDS aperture: low 32 bits = byte offset
        uint64_t ga = (uint64_t)(uintptr_t)(out + (size_t)c * VOX);
        // D# group 0: count=1 (valid), lds_addr, 57-bit global_addr, type=2 (image)
        u32x4 g0 = { 1u,
                     lds_base,
                     (uint32_t)ga,
                     (uint32_t)((ga >> 32) & 0x1FFFFFFu) | (2u << 30) };
        // D# group 1: data_size=4B; tensor_dim0=tile_dim0=32768; tensor_dim1=tile_dim1=1; stride0=32768
        i32x8 g1 = { (int)0x00020000,          // data_size=2 (4 bytes), no multicast/pad/iterate
                     (int)0x80000000,          // tensor_dim0[15:0]=32768 in bits[31:16]
                     (int)(1 << 16),           // tensor_dim0[31:16]=0, tensor_dim1[15:0]=1
                     (int)0x80000000,          // tensor_dim1[31:16]=0, tile_dim0=32768
                     1,                        // tile_dim1=1, tile_dim2=0
                     VOX,                      // tensor_dim0_stride[31:0]
                     0, 0 };                   // stride0 hi / tensor_dim1_stride = 0
        i32x4 z4 = { 0, 0, 0, 0 };
#if defined(__clang_major__) && (__clang_major__ >= 23)
        i32x8 z8 = { 0, 0, 0, 0, 0, 0, 0, 0 };
        __builtin_amdgcn_tensor_store_from_lds(g0, g1, z4, z4, z8, 0);
#else
        __builtin_amdgcn_tensor_store_from_lds(g0, g1, z4, z4, 0);
#endif
        __builtin_amdgcn_s_wait_tensorcnt(0); // hold wave (and WG's LDS) until TDM drains
    }
#else
    // Fallback: vectorized streaming stores
    float4* o4 = (float4*)(out + (size_t)c * VOX);
    #pragma unroll
    for (int i = 0; i < VOX / 4 / NTHREADS; ++i)
        o4[tid + i * NTHREADS] = g4[tid + i * NTHREADS];
#endif
}

extern "C" void kernel_launch(void* const* d_in, const int* in_sizes, int n_in,
                              void* d_out, int out_size, void* d_ws, size_t ws_size,
                              hipStream_t stream)
{
    const float* data   = (const float*)d_in[0];   // (N, 5) float32
    const int*   clusts = (const int*)d_in[1];     // (C, 4096) int32
    float*       out    = (float*)d_out;           // (C, 32, 32, 32) float32

    const int C = in_sizes[1] / PPC;               // 256 clusters
    clustcnn_voxelize_kernel<<<C, NTHREADS, VOX * sizeof(float), stream>>>(data, clusts, out);
}